// SpatialAttention_11948599017522
// MI455X (gfx1250) — compile-verified
//
#include <hip/hip_runtime.h>
#include <hip/hip_bf16.h>

// ---------------------------------------------------------------------------
// SpatialAttention for MI455X (gfx1250), fp32 WMMA path.
//   B=32, N=512, C=128, H=4, R=17, DH=32
// Stage 1: pre_kv[b,m,r,c] = sum_n x[b,n,c]*assignment[m,n,r]   (36.5 GFLOP)
// Stage 2: fused  kv-projection + masked softmax + out-projection (19.3 GFLOP)
// pre_kv lives in d_ws (142.6 MB, fits the 192 MB L2).
// ---------------------------------------------------------------------------

typedef float v2f __attribute__((ext_vector_type(2)));
typedef float v8f __attribute__((ext_vector_type(8)));

__device__ __forceinline__ v8f wmma4(v2f a, v2f b, v8f c) {
  // D = A(16x4,f32) * B(4x16,f32) + C(16x16,f32)
  return __builtin_amdgcn_wmma_f32_16x16x4_f32(false, a, false, b,
                                               (short)0, c, false, false);
}

#define B_  32
#define N_  512
#define C_  128
#define H_  4
#define R_  17
#define DH_ 32
#define SCALE_ 0.17677669529663687f  // 1/sqrt(32)

// ---------------------------------------------------------------------------
// Kernel 1: pre_kv GEMM.  grid = (N/64, R, B), block = 256 (8 wave32).
// Per workgroup: OUT[64 m x 128 c] = ASG[64 m x 512 n (r fixed)] * X[512 n x 128 c]
// ---------------------------------------------------------------------------
__global__ __launch_bounds__(256)
void prekv_gemm_kernel(const float* __restrict__ x,
                       const float* __restrict__ asg,
                       float* __restrict__ pre_kv) {
  __shared__ float lx[32][128];  // X chunk  [n][c], conflict-free B-frag reads
  __shared__ float la[32][65];   // ASG chunk transposed [n][m], +1 pad

  const int tid  = threadIdx.x;
  const int lane = tid & 31;
  const int w    = tid >> 5;
  const int l15  = lane & 15;
  const int off  = (lane >> 4) << 1;  // 0 (lanes 0-15) or 2 (lanes 16-31)
  const int hi8  = (lane >> 4) << 3;

  const int m0 = blockIdx.x * 64;
  const int r  = blockIdx.y;
  const int b  = blockIdx.z;

  const int mrow = (w >> 1) << 4;  // wave's 16-row m strip: 0,16,32,48
  const int cb   = (w & 1) << 6;   // wave's 64-col c strip: 0 or 64

  v8f acc[4] = {};

  const float* xb = x + (size_t)b * N_ * C_;
  const float* ab = asg + (size_t)m0 * (N_ * R_) + r;

  for (int n0 = 0; n0 < N_; n0 += 32) {
    __syncthreads();
    // stage X[n0..n0+32, 0..128): 1024 float4, fully coalesced
    {
      const float4* xs = (const float4*)(xb + (size_t)n0 * C_);
      float4* ld = (float4*)(&lx[0][0]);
#pragma unroll
      for (int e = 0; e < 4; ++e) ld[tid + 256 * e] = xs[tid + 256 * e];
    }
    // stage ASG[m0..m0+64, n0..n0+32, r] transposed into la[n][m]
#pragma unroll
    for (int e = 0; e < 8; ++e) {
      int idx = tid + 256 * e;   // 0..2047
      int n = idx & 31;
      int m = idx >> 5;
      la[n][m] = ab[(size_t)m * (N_ * R_) + (size_t)(n0 + n) * R_];
    }
    __syncthreads();
#pragma unroll
    for (int k = 0; k < 8; ++k) {
      const int k4 = k << 2;
      v2f a;
      a.x = la[k4 + off][mrow + l15];
      a.y = la[k4 + off + 1][mrow + l15];
#pragma unroll
      for (int j = 0; j < 4; ++j) {
        v2f bf;
        bf.x = lx[k4 + off][cb + (j << 4) + l15];
        bf.y = lx[k4 + off + 1][cb + (j << 4) + l15];
        acc[j] = wmma4(a, bf, acc[j]);
      }
    }
  }
  // store: D layout -> VGPR v, lanes0-15: M=v, lanes16-31: M=v+8
#pragma unroll
  for (int j = 0; j < 4; ++j) {
    const int c = cb + (j << 4) + l15;
#pragma unroll
    for (int v = 0; v < 8; ++v) {
      const int m = m0 + mrow + v + hi8;
      pre_kv[(((size_t)b * N_ + m) * R_ + r) * C_ + c] = acc[j][v];
    }
  }
}

// ---------------------------------------------------------------------------
// Kernel 2: fused q-proj + kv-proj (chunked) + masked softmax + out-proj.
// grid = (N/16, B), block = 256 (8 wave32). Each wg owns 16 output rows.
// pre_kv tile for this wg: 272 rows (m_local*17 + r) x 128, contiguous in ws.
// ---------------------------------------------------------------------------
__global__ __launch_bounds__(256)
void attn_fused_kernel(const float* __restrict__ x,
                       const float* __restrict__ prekv,
                       const unsigned char* __restrict__ mask,  // bool8 (N,R)
                       const float* __restrict__ Wq,
                       const float* __restrict__ Wkv,
                       const float* __restrict__ rel_bias,      // (H,1,R)
                       const float* __restrict__ Wp,
                       const float* __restrict__ bp,
                       float* __restrict__ out) {
  __shared__ float qt[16][132];       // q tile [m][h*32+d], pad 4
  __shared__ float chb[16][132];      // chunk buffer (X tile / k chunk / v chunk)
  __shared__ float sc[16][4][32];     // scores -> attn
  __shared__ float ctxs[16][132];     // context before out-projection

  const int tid  = threadIdx.x;
  const int lane = tid & 31;
  const int w    = tid >> 5;
  const int l15  = lane & 15;
  const int off  = (lane >> 4) << 1;
  const int hi8  = (lane >> 4) << 3;
  const int jb   = w << 4;            // wave's 16-wide output-column strip

  const int m0 = blockIdx.x * 16;
  const int b  = blockIdx.y;

  // ---- phase 0: stage X tile (16x128) into chb -----------------------------
#pragma unroll
  for (int e = 0; e < 8; ++e) {
    int idx = tid + (e << 8);  // 0..2047
    int m = idx >> 7, c = idx & 127;
    chb[m][c] = x[((size_t)b * N_ + m0 + m) * C_ + c];
  }
  __syncthreads();

  // ---- phase 1: q = Xtile @ Wq^T  (wave w -> columns jb..jb+15) ------------
  {
    v8f acc = {};
#pragma unroll
    for (int k = 0; k < 32; ++k) {
      const int k4 = k << 2;
      v2f a;
      a.x = chb[l15][k4 + off];
      a.y = chb[l15][k4 + off + 1];
      const float* wr = Wq + (size_t)(jb + l15) * C_ + k4 + off;
      v2f bf; bf.x = wr[0]; bf.y = wr[1];
      acc = wmma4(a, bf, acc);
    }
#pragma unroll
    for (int v = 0; v < 8; ++v) qt[v + hi8][jb + l15] = acc[v];
  }
  __syncthreads();

  const float* pkbase = prekv + (size_t)(b * N_ + m0) * R_ * C_;  // 272 x 128

  // ---- pass 1: k chunks -> scores ------------------------------------------
  for (int t = 0; t < 17; ++t) {
    {
      v8f acc = {};
#pragma unroll
      for (int k = 0; k < 32; ++k) {
        const int k4 = k << 2;
        const float* ar = pkbase + (size_t)((t << 4) + l15) * C_ + k4 + off;
        v2f a; a.x = ar[0]; a.y = ar[1];
        const float* wr = Wkv + (size_t)(jb + l15) * C_ + k4 + off;  // K half
        v2f bf; bf.x = wr[0]; bf.y = wr[1];
        acc = wmma4(a, bf, acc);
      }
      __syncthreads();  // previous readers of chb are done
#pragma unroll
      for (int v = 0; v < 8; ++v) chb[v + hi8][jb + l15] = acc[v];
    }
    __syncthreads();
    if (tid < 64) {  // 16 rows x 4 heads
      const int i = tid >> 2, h = tid & 3;
      const int g = (t << 4) + i;      // global row in 272-row tile
      const int m = g / 17, r = g % 17;
      float s = 0.f;
#pragma unroll
      for (int d = 0; d < 32; ++d) s += qt[m][(h << 5) + d] * chb[i][(h << 5) + d];
      sc[m][h][r] = s * SCALE_ + rel_bias[h * R_ + r];
    }
  }
  __syncthreads();

  // ---- masked softmax over r (one thread per (m,h)) ------------------------
  if (tid < 64) {
    const int m = tid >> 2, h = tid & 3;
    float vals[R_];
    float mx = -3.0e38f;
#pragma unroll
    for (int r = 0; r < R_; ++r) {
      const bool msk = mask[(size_t)(m0 + m) * R_ + r] != 0;
      const float s = msk ? -3.0e38f : sc[m][h][r];
      vals[r] = s;
      mx = fmaxf(mx, s);
    }
    float sum = 0.f;
#pragma unroll
    for (int r = 0; r < R_; ++r) {
      const float e = (vals[r] <= -1.0e38f) ? 0.f : __expf(vals[r] - mx);
      vals[r] = e;
      sum += e;
    }
    const float inv = 1.f / sum;  // mask[:,0]==False guarantees sum > 0
#pragma unroll
    for (int r = 0; r < R_; ++r) sc[m][h][r] = vals[r] * inv;
  }

  float ctx[8];
#pragma unroll
  for (int e = 0; e < 8; ++e) ctx[e] = 0.f;
  __syncthreads();

  // ---- pass 2: v chunks -> context -----------------------------------------
  for (int t = 0; t < 17; ++t) {
    {
      v8f acc = {};
#pragma unroll
      for (int k = 0; k < 32; ++k) {
        const int k4 = k << 2;
        const float* ar = pkbase + (size_t)((t << 4) + l15) * C_ + k4 + off;
        v2f a; a.x = ar[0]; a.y = ar[1];
        const float* wr = Wkv + (size_t)(C_ + jb + l15) * C_ + k4 + off;  // V half
        v2f bf; bf.x = wr[0]; bf.y = wr[1];
        acc = wmma4(a, bf, acc);
      }
      __syncthreads();
#pragma unroll
      for (int v = 0; v < 8; ++v) chb[v + hi8][jb + l15] = acc[v];
    }
    __syncthreads();
    // ctx[m][c] += attn[m, c/32, r] * v_chunk ; each thread owns 8 elements
#pragma unroll
    for (int e = 0; e < 8; ++e) {
      const int idx = tid + (e << 8);
      const int m = idx >> 7, c = idx & 127, h = c >> 5;
      float a = ctx[e];
#pragma unroll
      for (int i = 0; i < 16; ++i) {
        const int g = (t << 4) + i;
        const int mi = g / 17, ri = g % 17;
        if (mi == m) a += sc[m][h][ri] * chb[i][c];
      }
      ctx[e] = a;
    }
  }
  __syncthreads();
#pragma unroll
  for (int e = 0; e < 8; ++e) {
    const int idx = tid + (e << 8);
    ctxs[idx >> 7][idx & 127] = ctx[e];
  }
  __syncthreads();

  // ---- phase 3: out = ctx @ Wp^T + bp --------------------------------------
  {
    const float bpv = bp[jb + l15];
    v8f acc;
#pragma unroll
    for (int v = 0; v < 8; ++v) acc[v] = bpv;
#pragma unroll
    for (int k = 0; k < 32; ++k) {
      const int k4 = k << 2;
      v2f a;
      a.x = ctxs[l15][k4 + off];
      a.y = ctxs[l15][k4 + off + 1];
      const float* wr = Wp + (size_t)(jb + l15) * C_ + k4 + off;
      v2f bf; bf.x = wr[0]; bf.y = wr[1];
      acc = wmma4(a, bf, acc);
    }
#pragma unroll
    for (int v = 0; v < 8; ++v) {
      const int m = m0 + v + hi8;
      out[((size_t)b * N_ + m) * C_ + jb + l15] = acc[v];
    }
  }
}

// ---------------------------------------------------------------------------
extern "C" void kernel_launch(void* const* d_in, const int* in_sizes, int n_in,
                              void* d_out, int out_size, void* d_ws, size_t ws_size,
                              hipStream_t stream) {
  const float*         x    = (const float*)d_in[0];
  const float*         asg  = (const float*)d_in[1];
  const unsigned char* mask = (const unsigned char*)d_in[2];  // jax bool = 1 byte
  const float*         Wq   = (const float*)d_in[3];
  const float*         Wkv  = (const float*)d_in[4];
  const float*         rb   = (const float*)d_in[5];
  const float*         Wp   = (const float*)d_in[6];
  const float*         bp   = (const float*)d_in[7];
  float* prekv = (float*)d_ws;           // B*N*R*C floats = 142.6 MB scratch
  float* out   = (float*)d_out;

  // Stage 1: pre_kv = einsum('bnc,mnr->bmrc')
  prekv_gemm_kernel<<<dim3(N_ / 64, R_, B_), 256, 0, stream>>>(x, asg, prekv);
  // Stage 2: fused projections + attention
  attn_fused_kernel<<<dim3(N_ / 16, B_), 256, 0, stream>>>(
      x, prekv, mask, Wq, Wkv, rb, Wp, bp, out);
}